// T_SAGE_70746701300059
// MI455X (gfx1250) — compile-verified
//
#include <hip/hip_runtime.h>
#include <stdint.h>

typedef __attribute__((ext_vector_type(2))) float v2f;
typedef __attribute__((ext_vector_type(8))) float v8f;

#define DD 128  // feature dim

// LDS strides chosen for conflict-free access on 64 banks:
//  A rows: stride 132 (132 % 64 = 4; 16 rows * 4 -> 16 distinct banks)
//  W rows: stride 36  (gcd(36,64)=4, 9 coprime 16 -> 16 distinct banks)
#define ASTRIDE 132
#define WSTRIDE 36
#define KSLICE  32

// ---------------------------------------------------------------- utilities

__device__ __forceinline__ uint32_t rotl32(uint32_t x, int r) {
  return (x << r) | (x >> (32 - r));
}

// Exact JAX threefry-2x32 dropout mask for jax.random.bernoulli(key(1), 0.5, (rows,128)).
// key(1) -> key data (0,1). random_bits over n elements: counters split in halves,
// element idx<n/2 takes lane-0 output of pair (idx, idx+n/2), else lane-1.
__device__ bool jax_keep(uint32_t idx, uint32_t n) {
  const uint32_t half = n >> 1;  // n = rows*128 is even
  uint32_t i, pos;
  if (idx < half) { i = idx;        pos = 0; }
  else            { i = idx - half; pos = 1; }
  uint32_t x0 = i, x1 = i + half;
  const uint32_t ks0 = 0u, ks1 = 1u, ks2 = 0x1BD11BDBu;  // 0^1^0x1BD11BDA
  x0 += ks0; x1 += ks1;
  const int r1[4] = {13, 15, 26, 6};
  const int r2[4] = {17, 29, 16, 24};
#pragma unroll
  for (int g = 0; g < 5; ++g) {
#pragma unroll
    for (int k = 0; k < 4; ++k) {
      int r = ((g & 1) == 0) ? r1[k] : r2[k];
      x0 += x1; x1 = rotl32(x1, r); x1 ^= x0;
    }
    switch (g) {
      case 0: x0 += ks1; x1 += ks2 + 1u; break;
      case 1: x0 += ks2; x1 += ks0 + 2u; break;
      case 2: x0 += ks0; x1 += ks1 + 3u; break;
      case 3: x0 += ks1; x1 += ks2 + 4u; break;
      case 4: x0 += ks2; x1 += ks0 + 5u; break;
    }
  }
  uint32_t bits = (pos == 0) ? x0 : x1;
  float u = __uint_as_float((bits >> 9) | 0x3f800000u) - 1.0f;
  return u < 0.5f;
}

// ---------------------------------------------------------------- kernels

__global__ void zero_kernel(float* p, long n) {
  long i = (long)blockIdx.x * blockDim.x + threadIdx.x;
  long stride = (long)gridDim.x * blockDim.x;
  for (; i < n; i += stride) p[i] = 0.0f;
}

// One edge per 128 lanes; float atomics into agg[dst], deg[dst].
__global__ void scatter_kernel(const float* __restrict__ x,
                               const long long* __restrict__ eidx,  // [2, E]
                               long E,
                               const int* __restrict__ limit_p,     // only dst < limit matter
                               float* __restrict__ agg,
                               float* __restrict__ deg) {
  const int lim = *limit_p;
  const long e = (long)blockIdx.x * 2 + (threadIdx.x >> 7);
  const int lane = threadIdx.x & 127;
  if (e >= E) return;
  const long long src = eidx[e];
  const long long dst = eidx[E + e];
  if (dst >= lim) return;
  atomicAdd(&agg[(long)dst * DD + lane], x[(long)src * DD + lane]);
  if (lane == 0) atomicAdd(&deg[dst], 1.0f);
}

// out[m][n] = sum_k (agg[m][k]/max(deg,1)) * Wl[n][k] + sum_k xin[m][k] * Wr[n][k] + b[n]
// Block = 256 threads = 8 waves; block -> 16-row strip; wave w -> columns [16w,16w+16).
// A strips (deg-normalized agg + xin) staged once in LDS; weights staged in
// 32-wide K slices (coalesced global loads shared by all 8 waves).
// Inner loop: ds_load_b64 + V_WMMA_F32_16X16X4_F32.
template <bool RELU_DROP>
__global__ void sage_gemm_kernel(const float* __restrict__ aggm,  // [*, D]
                                 const float* __restrict__ deg,   // [*]
                                 const float* __restrict__ xin,   // [*, D]
                                 const float* __restrict__ Wl,    // [D, D] (out,in)
                                 const float* __restrict__ Wr,    // [D, D]
                                 const float* __restrict__ bias,  // [D]
                                 const int* __restrict__ rows_p,  // valid rows
                                 float* __restrict__ out) {       // [*, D]
  __shared__ float lds_a[16 * ASTRIDE];
  __shared__ float lds_x[16 * ASTRIDE];
  __shared__ float lds_wl[DD * WSTRIDE];
  __shared__ float lds_wr[DD * WSTRIDE];

  const int rows = *rows_p;
  const int m0 = blockIdx.x * 16;
  if (m0 >= rows) return;  // uniform branch: whole block exits together

  const int tid = threadIdx.x;
  const int wave = tid >> 5;   // 0..7
  const int lane = tid & 31;
  const int n0 = wave * 16;
  const int half = lane >> 4;  // K-pair select for A/B
  const int l15 = lane & 15;
  const int bcol = n0 + l15;   // B/D column owned by this lane

  // Stage A strips (16 rows x 128) with degree normalization applied.
  // Reads past `rows` stay inside the padded ws regions (garbage rows are
  // never stored). Coalesced: consecutive tids -> consecutive columns.
  for (int s = tid; s < 16 * DD; s += 256) {
    const int row = s >> 7;
    const int col = s & (DD - 1);
    const float rd = 1.0f / fmaxf(deg[m0 + row], 1.0f);
    lds_a[row * ASTRIDE + col] = aggm[(long)(m0 + row) * DD + col] * rd;
    lds_x[row * ASTRIDE + col] = xin[(long)(m0 + row) * DD + col];
  }

  v8f acc = {};
#pragma unroll
  for (int kb = 0; kb < DD / KSLICE; ++kb) {
    __syncthreads();  // previous slice fully consumed / A staging visible
    // Stage Wl/Wr K-slices: [n=0..127][kk=0..31]; coalesced 32-float row chunks.
    for (int s = tid; s < DD * KSLICE; s += 256) {
      const int n = s >> 5;
      const int kk = s & (KSLICE - 1);
      lds_wl[n * WSTRIDE + kk] = Wl[(long)n * DD + kb * KSLICE + kk];
      lds_wr[n * WSTRIDE + kk] = Wr[(long)n * DD + kb * KSLICE + kk];
    }
    __syncthreads();

#pragma unroll
    for (int k8 = 0; k8 < KSLICE / 4; ++k8) {
      const int k = kb * KSLICE + k8 * 4;
      const int ka = k + half * 2;            // global K index for A
      const int kl = k8 * 4 + half * 2;       // local K index for W slice
      v2f a1 = { lds_a[l15 * ASTRIDE + ka], lds_a[l15 * ASTRIDE + ka + 1] };
      v2f b1 = { lds_wl[bcol * WSTRIDE + kl], lds_wl[bcol * WSTRIDE + kl + 1] };
      acc = __builtin_amdgcn_wmma_f32_16x16x4_f32(false, a1, false, b1,
                                                  (short)0, acc, false, false);
      v2f a2 = { lds_x[l15 * ASTRIDE + ka], lds_x[l15 * ASTRIDE + ka + 1] };
      v2f b2 = { lds_wr[bcol * WSTRIDE + kl], lds_wr[bcol * WSTRIDE + kl + 1] };
      acc = __builtin_amdgcn_wmma_f32_16x16x4_f32(false, a2, false, b2,
                                                  (short)0, acc, false, false);
    }
  }

  const float bv = bias[bcol];
  const int mbase = m0 + half * 8;  // C/D: VGPR r holds M = r + 8*(lane>=16)
#pragma unroll
  for (int r = 0; r < 8; ++r) {
    const int m = mbase + r;
    if (m >= rows) continue;  // predicated store only; WMMA already done
    float v = acc[r] + bv;
    if (RELU_DROP) {
      v = fmaxf(v, 0.0f);
      const uint32_t idx = (uint32_t)m * DD + (uint32_t)bcol;
      v = jax_keep(idx, (uint32_t)rows * DD) ? v * 2.0f : 0.0f;
    }
    out[(long)m * DD + bcol] = v;
  }
}

// ---------------------------------------------------------------- launch

extern "C" void kernel_launch(void* const* d_in, const int* in_sizes, int n_in,
                              void* d_out, int out_size, void* d_ws, size_t ws_size,
                              hipStream_t stream) {
  const float*     feat  = (const float*)d_in[0];
  const long long* t_adj = (const long long*)d_in[1];
  const long long* n_adj = (const long long*)d_in[2];
  const int*       ip    = (const int*)d_in[3];   // i (device scalar)
  const int*       jp    = (const int*)d_in[4];   // j (device scalar)
  const float*     W1l   = (const float*)d_in[5];
  const float*     b1    = (const float*)d_in[6];
  const float*     W1r   = (const float*)d_in[7];
  const float*     W2l   = (const float*)d_in[8];
  const float*     b2    = (const float*)d_in[9];
  const float*     W2r   = (const float*)d_in[10];
  float*           out   = (float*)d_out;

  const long N  = in_sizes[0] / DD;   // 50000
  const long E1 = in_sizes[1] / 2;    // 600000
  const long E2 = in_sizes[2] / 2;    // 300000
  const long I  = out_size / DD;      // 12500 (= i)

  // Workspace layout (floats). j is only known on-device, so layer-1 regions
  // are sized with the safe upper bound N (j <= N); layer-2 with I (+ strip pad).
  float* ws = (float*)d_ws;
  long off = 0;
  float* agg1 = ws + off; off += N * DD;
  float* deg1 = ws + off; off += N;
  float* x1   = ws + off; off += N * DD;
  float* agg2 = ws + off; off += (I + 16) * DD;
  float* deg2 = ws + off; off += (I + 16);
  const long total = off;

  zero_kernel<<<2048, 256, 0, stream>>>(ws, total);

  // Layer 1: aggregate feat over t_adj (dst < j), then WMMA GEMMs + relu + dropout.
  scatter_kernel<<<(unsigned)((E1 + 1) / 2), 256, 0, stream>>>(feat, t_adj, E1, jp, agg1, deg1);
  sage_gemm_kernel<true><<<(unsigned)((N + 15) / 16), 256, 0, stream>>>(
      agg1, deg1, feat, W1l, W1r, b1, jp, x1);

  // Layer 2: aggregate x1 over n_adj (dst < i), then WMMA GEMMs -> d_out.
  scatter_kernel<<<(unsigned)((E2 + 1) / 2), 256, 0, stream>>>(x1, n_adj, E2, ip, agg2, deg2);
  sage_gemm_kernel<false><<<(unsigned)((I + 15) / 16), 256, 0, stream>>>(
      agg2, deg2, x1, W2l, W2r, b2, ip, out);
}